// WeightedAverage_34084860461733
// MI455X (gfx1250) — compile-verified
//
#include <hip/hip_runtime.h>
#include <cstdint>

// Weighted local-color average (Lab): memory-bound 3x3 data-dependent stencil.
// CDNA5 path: Tensor Data Mover bulk-copies each block's haloed [3][18][66]
// tile into LDS (tensor_load_to_lds + s_wait_tensorcnt), compute from LDS.

#define TW 64
#define TH 16
#define LW (TW + 2)          // 66
#define LH (TH + 2)          // 18
#define IMG_H 512
#define IMG_W 512
#define PLANE (IMG_H * IMG_W)

typedef unsigned int u32x4 __attribute__((ext_vector_type(4)));
typedef int          i32x8 __attribute__((ext_vector_type(8)));
typedef int          i32x4 __attribute__((ext_vector_type(4)));

__global__ __launch_bounds__(TW * TH)
void wavg_lab_kernel(const float* __restrict__ x,
                     const int*   __restrict__ alpha_p,
                     float*       __restrict__ out)
{
    __shared__ float lds[3 * LH * LW];

    const int bx = blockIdx.x, by = blockIdx.y, n = blockIdx.z;
    const int tx = threadIdx.x, ty = threadIdx.y;
    const int tid = ty * TW + tx;

    const int gx0 = bx * TW - 1;           // tile top-left incl. halo
    const int gy0 = by * TH - 1;
    const float* img = x + (size_t)n * 3 * PLANE;

    const bool interior = (gx0 >= 0) && (gy0 >= 0) &&
                          (gx0 + LW <= IMG_W) && (gy0 + LH <= IMG_H);

#if defined(__gfx1250__) && __has_builtin(__builtin_amdgcn_tensor_load_to_lds)
    if (interior) {
        if (tid == 0) {
            // ---- Tensor DMA descriptor (D#): 3D tile [Z=3][Y=LH][X=LW] ----
            unsigned long long ga =
                (unsigned long long)(uintptr_t)(img + (size_t)gy0 * IMG_W + gx0);
            unsigned int lds_addr = (unsigned int)(uintptr_t)(&lds[0]);

            u32x4 g0;
            g0.x = 1u;                                   // count=1 (valid), user mode
            g0.y = lds_addr;                             // lds_addr (bytes)
            g0.z = (unsigned int)(ga & 0xffffffffu);     // global_addr[31:0]
            g0.w = (unsigned int)((ga >> 32) & 0x01ffffffu) | 0x80000000u; // [56:32] | type=2

            i32x8 g1;
            g1[0] = (int)(2u << 16);                     // data_size=4B; no mcast/pad/iter
            g1[1] = (int)((unsigned)LW << 16);           // tensor_dim0[15:0]
            g1[2] = (int)((unsigned)LH << 16);           // dim0[31:16]=0 | tensor_dim1[15:0]
            g1[3] = (int)((unsigned)LW << 16);           // dim1[31:16]=0 | tile_dim0=LW
            g1[4] = (int)((unsigned)LH | (3u << 16));    // tile_dim1=LH | tile_dim2=3
            g1[5] = IMG_W;                               // tensor_dim0_stride[31:0] (elems)
            g1[6] = (int)(((unsigned)PLANE & 0xffffu) << 16); // d0s[47:32]=0 | d1s[15:0]
            g1[7] = (int)((unsigned)PLANE >> 16);        // tensor_dim1_stride[47:16]

            i32x4 g2;                                    // tensor_dim2=3, rest unused
            g2[0] = 3; g2[1] = 0; g2[2] = 0; g2[3] = 0;
            i32x4 g3 = {0, 0, 0, 0};
            i32x8 g4 = {0, 0, 0, 0, 0, 0, 0, 0};         // extra group (clang-23 form)

            __builtin_amdgcn_tensor_load_to_lds(g0, g1, g2, g3, g4, 0);
            __builtin_amdgcn_s_wait_tensorcnt(0);        // s_wait_tensorcnt 0x0
        }
    } else
#endif
    {
        // Border tiles (and fallback): per-lane loads with zero padding.
        for (int idx = tid; idx < 3 * LH * LW; idx += TW * TH) {
            int c   = idx / (LH * LW);
            int rem = idx - c * (LH * LW);
            int r   = rem / LW;
            int cc  = rem - r * LW;
            int gy = gy0 + r, gx = gx0 + cc;
            float v = 0.0f;
            if (gy >= 0 && gy < IMG_H && gx >= 0 && gx < IMG_W)
                v = img[(size_t)c * PLANE + (size_t)gy * IMG_W + gx];
            lds[idx] = v;
        }
    }
    __syncthreads();

    // ---- compute: softmax over 9 taps of (L - Lc)^2, weighted avg of a,b ----
    const float inv_alpha = 1.0f / (float)alpha_p[0];
    const float* Ls = lds;
    const float* As = lds + LH * LW;
    const float* Bs = lds + 2 * LH * LW;

    const int base = ty * LW + tx;                 // top-left of this pixel's 3x3
    const float lc = Ls[base + LW + 1];

    float wsum = 0.f, wa = 0.f, wb = 0.f;
#pragma unroll
    for (int i = 0; i < 3; ++i) {
#pragma unroll
        for (int j = 0; j < 3; ++j) {
            const int o = base + i * LW + j;
            const float d = Ls[o] - lc;
            const float w = __expf(-d * d * inv_alpha);  // max-shift == 0 exactly
            wsum += w;
            wa   += w * As[o];
            wb   += w * Bs[o];
        }
    }

    const int gx = bx * TW + tx;
    const int gy = by * TH + ty;
    const size_t opix = (size_t)gy * IMG_W + gx;
    float* outn = out + (size_t)n * 2 * PLANE;
    const float inv = 1.0f / wsum;
    outn[opix]          = wa * inv;   // wa -> channel 0
    outn[PLANE + opix]  = wb * inv;   // wb -> channel 1
}

extern "C" void kernel_launch(void* const* d_in, const int* in_sizes, int n_in,
                              void* d_out, int out_size, void* d_ws, size_t ws_size,
                              hipStream_t stream) {
    const float* x     = (const float*)d_in[0];   // x_lab [N,3,512,512] f32
    const int*   alpha = (const int*)  d_in[2];   // alpha (scalar int)
    float*       out   = (float*)d_out;           // [N,2,512,512] f32

    const int N = in_sizes[0] / (3 * PLANE);
    dim3 grid(IMG_W / TW, IMG_H / TH, N);
    dim3 block(TW, TH, 1);
    hipLaunchKernelGGL(wavg_lab_kernel, grid, block, 0, stream, x, alpha, out);
}